// GCN_Hidden_Anchored_29643864277064
// MI455X (gfx1250) — compile-verified
//
#include <hip/hip_runtime.h>

typedef float v2f __attribute__((ext_vector_type(2)));
typedef float v8f __attribute__((ext_vector_type(8)));

// ---------------- degree / norm ----------------
__global__ void deg_count(const int* __restrict__ dst, float* __restrict__ deg, int nE) {
  int e = blockIdx.x * blockDim.x + threadIdx.x;
  if (e < nE) atomicAdd(&deg[dst[e]], 1.0f);
}

__global__ void deg_to_dinv(float* __restrict__ deg, int n) {
  int i = blockIdx.x * blockDim.x + threadIdx.x;
  if (i < n) deg[i] = rsqrtf(deg[i] + 1.0f);  // +1 self-loop; always > 0
}

// ---------------- fp32 WMMA GEMM: C[M,NOUT] = A[M,K] @ B[K,NOUT] ----------------
// One wave computes one 16x16 output tile using V_WMMA_F32_16X16X4_F32, K stepped by 4.
// A layout (ISA 7.12.2, 32-bit A 16x4): lanes 0-15 rows M, VGPR0=K0/VGPR1=K1;
// lanes 16-31 same rows, VGPR0=K2/VGPR1=K3. B/C/D row-striped across lanes per VGPR.
template <int K, int NOUT>
__global__ void wmma_gemm_f32(const float* __restrict__ A, const float* __restrict__ B,
                              float* __restrict__ C, int M) {
  int wave = blockIdx.x * (blockDim.x >> 5) + (threadIdx.x >> 5);
  int lane = threadIdx.x & 31;
  const int tiles_n = NOUT >> 4;
  int tm = wave / tiles_n;
  int tn = wave - tm * tiles_n;
  if (tm * 16 >= M) return;  // wave-uniform guard: EXEC stays all-1s for WMMA

  const int half = lane >> 4;   // 0: lanes 0-15, 1: lanes 16-31
  const int l15  = lane & 15;
  const float* arow = A + (size_t)(tm * 16 + l15) * K + half * 2;
  const float* bcol = B + (size_t)(half * 2) * NOUT + tn * 16 + l15;

  v8f acc = {};
#pragma unroll 4
  for (int k = 0; k < K; k += 4) {
    v2f a = *(const v2f*)(arow + k);              // K = k+2*half, k+1+2*half
    const float* bp = bcol + (size_t)k * NOUT;
    v2f b;
    b.x = bp[0];                                   // row K = k+2*half
    b.y = bp[NOUT];                                // row K = k+1+2*half
    acc = __builtin_amdgcn_wmma_f32_16x16x4_f32(
        /*neg_a=*/false, a, /*neg_b=*/false, b,
        /*c_mod=*/(short)0, acc, /*reuse_a=*/false, /*reuse_b=*/false);
  }
  // C layout: VGPR r -> row tm*16 + r + 8*half, col tn*16 + l15
  float* crow = C + (size_t)(tm * 16 + half * 8) * NOUT + tn * 16 + l15;
#pragma unroll
  for (int r = 0; r < 8; ++r) crow[(size_t)r * NOUT] = acc[r];
}

// ---------------- self-loop init: out = bias + dinv[i]^2 * h ----------------
template <int F>
__global__ void self_init(const float* __restrict__ h, const float* __restrict__ dinv,
                          const float* __restrict__ bias, float* __restrict__ out, int n) {
  int gid = blockIdx.x * blockDim.x + threadIdx.x;
  int i = gid / F;
  int f = gid - i * F;
  if (i < n) {
    float w = dinv[i];
    out[gid] = bias[f] + w * w * h[gid];
  }
}

// ---------------- edge scatter: out[dst] += h[src] * dinv[src]*dinv[dst] ----------------
// F/4 threads per edge, float4 gather (coalesced; h fits in 192MB L2), f32 atomics.
template <int F>
__global__ void scatter_norm_add(const float* __restrict__ h, const int* __restrict__ src,
                                 const int* __restrict__ dst, const float* __restrict__ dinv,
                                 float* __restrict__ out, int nE) {
  const int TPE = F >> 2;
  int gid = blockIdx.x * blockDim.x + threadIdx.x;
  int e = gid / TPE;
  if (e >= nE) return;
  int c = (gid - e * TPE) << 2;
  int s = src[e], d = dst[e];
  float norm = dinv[s] * dinv[d];
  const float4 v = *(const float4*)(h + (size_t)s * F + c);
  float* o = out + (size_t)d * F + c;
  atomicAdd(o + 0, v.x * norm);
  atomicAdd(o + 1, v.y * norm);
  atomicAdd(o + 2, v.z * norm);
  atomicAdd(o + 3, v.w * norm);
}

// ---------------- relu + anchored concat: h2 = [relu(agg)-c, c], c = relu(agg)[perm] ----------------
__global__ void relu_build_h2(const float* __restrict__ agg, const int* __restrict__ perm,
                              float* __restrict__ h2, int n) {
  int gid = blockIdx.x * blockDim.x + threadIdx.x;
  int i = gid >> 7;          // HID = 128
  int f = gid & 127;
  if (i >= n) return;
  float hv = fmaxf(agg[gid], 0.0f);
  float cv = fmaxf(agg[(size_t)perm[i] * 128 + f], 0.0f);
  size_t o = (size_t)i * 256 + f;
  h2[o] = hv - cv;
  h2[o + 128] = cv;
}

extern "C" void kernel_launch(void* const* d_in, const int* in_sizes, int n_in,
                              void* d_out, int out_size, void* d_ws, size_t ws_size,
                              hipStream_t stream) {
  const float* x  = (const float*)d_in[0];
  const float* W1 = (const float*)d_in[1];
  const float* b1 = (const float*)d_in[2];
  const float* W2 = (const float*)d_in[3];
  const float* b2 = (const float*)d_in[4];
  const int* ei   = (const int*)d_in[5];
  const int* perm = (const int*)d_in[6];
  const int N = in_sizes[6];        // 100000
  const int E = in_sizes[5] / 2;    // 1600000
  const int* src = ei;              // edge_index[0]
  const int* dst = ei + E;          // edge_index[1]

  // workspace carve (fp32): dinv[N] | h1[N*128] | agg1[N*128] | h2[N*256]; h3 reuses h1
  float* dinv = (float*)d_ws;
  float* h1   = dinv + (size_t)((N + 63) & ~63);
  float* agg1 = h1 + (size_t)N * 128;
  float* h2   = agg1 + (size_t)N * 128;
  float* h3   = h1;  // h1 dead once h2 is built

  // 1) degree -> dinv
  hipMemsetAsync(dinv, 0, sizeof(float) * (size_t)N, stream);
  deg_count<<<(E + 255) / 256, 256, 0, stream>>>(dst, dinv, E);
  deg_to_dinv<<<(N + 255) / 256, 256, 0, stream>>>(dinv, N);

  // 2) h1 = x @ W1  (fp32 WMMA)
  {
    int tiles = ((N + 15) / 16) * (128 / 16);
    wmma_gemm_f32<256, 128><<<(tiles + 7) / 8, 256, 0, stream>>>(x, W1, h1, N);
  }
  // 3) agg1 = b1 + dinv^2 * h1 (self-loop term, atomic-free init)
  self_init<128><<<(N * 128 + 255) / 256, 256, 0, stream>>>(h1, dinv, b1, agg1, N);
  // 4) agg1[dst] += h1[src] * norm
  scatter_norm_add<128><<<(E * 32 + 255) / 256, 256, 0, stream>>>(h1, src, dst, dinv, agg1, E);
  // 5) h2 = [relu(agg1) - relu(agg1)[perm], relu(agg1)[perm]]
  relu_build_h2<<<(N * 128 + 255) / 256, 256, 0, stream>>>(agg1, perm, h2, N);

  // 6) h3 = h2 @ W2  (fp32 WMMA)
  {
    int tiles = ((N + 15) / 16) * (64 / 16);
    wmma_gemm_f32<256, 64><<<(tiles + 7) / 8, 256, 0, stream>>>(h2, W2, h3, N);
  }
  float* out = (float*)d_out;
  // 7) out = b2 + dinv^2 * h3
  self_init<64><<<(N * 64 + 255) / 256, 256, 0, stream>>>(h3, dinv, b2, out, N);
  // 8) out[dst] += h3[src] * norm
  scatter_norm_add<64><<<(E * 16 + 255) / 256, 256, 0, stream>>>(h3, src, dst, dinv, out, E);
}